// PixelDINOLoss_62036507623554
// MI455X (gfx1250) — compile-verified
//
#include <hip/hip_runtime.h>
#include <stdint.h>
#include <math.h>

// PixelDINO loss, MI455X (gfx1250, wave32).
// B=8, D=128, H=W=256. Memory-bound: 512 MB fp32 in, ~22us floor @ 23.3 TB/s.
// Two passes per pixel (norms, then softmax CE); pass 2 re-reads from L2.
// All five per-pixel reductions over D are done with v_wmma_f32_16x16x4_f32
// against an all-ones B matrix (rowsum-via-matrix): free under the memory
// roofline, fp32-exact layout, no shuffle trees.

#define BATCH 8
#define DCH   128
#define HWPIX 65536            // 256*256
#define NPIX  (BATCH * HWPIX)  // 524288
#define PPB   128              // pixels per block (8 waves * 16 pixels)
#define NBLK  (NPIX / PPB)     // 4096
#define BLKS_PER_IMG (HWPIX / PPB) // 512

typedef float v2f __attribute__((ext_vector_type(2)));
typedef float v8f __attribute__((ext_vector_type(8)));

// D += A(16x4) * ones(4x16): accumulates per-row (per-pixel) sums over the
// K=4 chunk, replicated across all 16 columns of the C/D tile.
__device__ inline v8f wmma_rowacc(v2f a, v8f c) {
  const v2f ones = {1.0f, 1.0f};
  return __builtin_amdgcn_wmma_f32_16x16x4_f32(
      /*neg_a=*/false, a, /*neg_b=*/false, ones,
      /*c_mod=*/(short)0, c, /*reuse_a=*/false, /*reuse_b=*/false);
}

__device__ inline float pick8(v8f a, int idx) {
  float r = a[0];
#pragma unroll
  for (int i = 1; i < 8; ++i) r = (idx == i) ? (float)a[i] : r;
  return r;
}

// C/D layout: VGPR r holds row M=r (lanes 0-15) and M=8+r (lanes 16-31).
// Return row p's value to every lane whose pixel index is p.
__device__ inline float row_val(v8f acc, int p) {
  float x = pick8(acc, p & 7);
  int src = p + ((p & 8) << 1);  // p<8 -> lane p (low half); p>=8 -> lane p+16
  return __shfl(x, src, 32);
}

__global__ __launch_bounds__(256) void dino_pixel_kernel(
    const float* __restrict__ S,   // student [B,D,H,W]
    const float* __restrict__ T,   // teacher [B,D,H,W]
    const float* __restrict__ C,   // center  [D]
    const uint8_t* __restrict__ M, // mask    [B,H,W] (jax bool = 1 byte)
    const float* __restrict__ O,   // original_x [B,1,H,W]
    float* __restrict__ ws)        // [0..NBLK): loss sums, [NBLK..2*NBLK): counts
{
  __shared__ float cS[DCH];
  __shared__ float redL[8];
  __shared__ float redC[8];

  const int tid = threadIdx.x;
  const int w   = tid >> 5;       // wave id in block
  const int L   = tid & 31;       // lane
  const int p   = L & 15;         // pixel within wave tile
  const int g   = L >> 4;         // K-half (A-matrix lanes 16-31 hold K=2,3)

  if (tid < DCH) cS[tid] = C[tid];
  __syncthreads();

  const int P  = blockIdx.x * PPB + w * 16 + p;  // global flat pixel in [B*H*W)
  const int b  = P >> 16;
  const int hw = P & (HWPIX - 1);
  const size_t pixb = ((size_t)b << 23) + (size_t)hw;  // (b, d=0, hw) element idx
  const float* Sp = S + pixb;
  const float* Tp = T + pixb;

  // ---- Pass 1: sum of squares for student and centered teacher ----
  v8f accSS = {};
  v8f accTT = {};
#pragma unroll 4
  for (int k = 0; k < 32; ++k) {
    const int d0 = k * 4 + g * 2;
    const size_t off = (size_t)d0 << 16;
    __builtin_prefetch((const void*)(Sp + off + ((size_t)4 << 16)), 0, 1);
    __builtin_prefetch((const void*)(Tp + off + ((size_t)4 << 16)), 0, 1);
    const float s0 = Sp[off];
    const float s1 = Sp[off + HWPIX];
    const float t0 = Tp[off] - cS[d0];
    const float t1 = Tp[off + HWPIX] - cS[d0 + 1];
    v2f ass = {s0 * s0, s1 * s1};
    v2f att = {t0 * t0, t1 * t1};
    accSS = wmma_rowacc(ass, accSS);
    accTT = wmma_rowacc(att, accTT);
  }

  const float ss = row_val(accSS, p);
  const float tt = row_val(accTT, p);
  const float inv_ns = 1.0f / fmaxf(sqrtf(ss), 1e-12f);
  const float inv_nt = 1.0f / fmaxf(sqrtf(tt), 1e-12f);
  const float as = inv_ns * 10.0f;  // 1/tau_s
  const float at = inv_nt * 25.0f;  // 1/tau_t
  // |s_hat/tau_s| <= 10, |t_hat/tau_t| <= 25 -> exp safe in fp32 without max-sub.

  // ---- Pass 2: Z_s, Z_t, cross term (re-read; L2-resident) ----
  v8f zs = {}, zt = {}, xc = {};
#pragma unroll 4
  for (int k = 0; k < 32; ++k) {
    const int d0 = k * 4 + g * 2;
    const size_t off = (size_t)d0 << 16;
    const float s0 = Sp[off];
    const float s1 = Sp[off + HWPIX];
    const float t0 = Tp[off] - cS[d0];
    const float t1 = Tp[off + HWPIX] - cS[d0 + 1];
    const float ls0 = s0 * as, ls1 = s1 * as;
    const float es0 = __expf(ls0), es1 = __expf(ls1);
    const float et0 = __expf(t0 * at), et1 = __expf(t1 * at);
    v2f azs = {es0, es1};
    v2f azt = {et0, et1};
    v2f axc = {et0 * ls0, et1 * ls1};
    zs = wmma_rowacc(azs, zs);
    zt = wmma_rowacc(azt, zt);
    xc = wmma_rowacc(axc, xc);
  }

  // loss = log(Z_s) - cross/Z_t, computed elementwise in C/D layout
  v8f lv;
#pragma unroll
  for (int i = 0; i < 8; ++i)
    lv[i] = __logf((float)zs[i]) - (float)xc[i] / (float)zt[i];
  const float loss_p = row_val(lv, p);

  // ---- masked accumulation (only g==0 lanes own a pixel's bookkeeping) ----
  float cl = 0.0f, cc = 0.0f;
  if (g == 0) {
    const float o = O[P];
    const bool valid = (o != 0.0f) && (M[P] == 0);
    cl = valid ? loss_p : 0.0f;
    cc = valid ? 1.0f : 0.0f;
  }
#pragma unroll
  for (int off = 16; off > 0; off >>= 1) {
    cl += __shfl_xor(cl, off, 32);
    cc += __shfl_xor(cc, off, 32);
  }
  if (L == 0) { redL[w] = cl; redC[w] = cc; }
  __syncthreads();
  if (tid == 0) {
    float a = 0.0f, d = 0.0f;
#pragma unroll
    for (int i = 0; i < 8; ++i) { a += redL[i]; d += redC[i]; }
    ws[blockIdx.x]        = a;   // deterministic per-block slot (no atomics)
    ws[NBLK + blockIdx.x] = d;
  }
}

__global__ void dino_finalize_kernel(const float* __restrict__ ws,
                                     float* __restrict__ out) {
  __shared__ float pi[BATCH];
  __shared__ float cz[BATCH];
  const int t = threadIdx.x;
  if (t < BATCH) {
    float s = 0.0f, c = 0.0f;
    for (int i = 0; i < BLKS_PER_IMG; ++i) {
      s += ws[t * BLKS_PER_IMG + i];
      c += ws[NBLK + t * BLKS_PER_IMG + i];
    }
    pi[t] = s / fmaxf(c, 1.0f);
    cz[t] = c;
  }
  __syncthreads();
  if (t == 0) {
    float tot = 0.0f, nnz = 0.0f, call = 0.0f;
    for (int i = 0; i < BATCH; ++i) {
      call += cz[i];
      if (cz[i] > 0.0f) { tot += pi[i]; nnz += 1.0f; }
    }
    out[0] = (call > 0.0f) ? (tot / fmaxf(nnz, 1.0f)) : 0.0f;
  }
}

extern "C" void kernel_launch(void* const* d_in, const int* in_sizes, int n_in,
                              void* d_out, int out_size, void* d_ws, size_t ws_size,
                              hipStream_t stream) {
  const float*   S = (const float*)d_in[0];    // student_feats [B,D,H,W]
  const float*   T = (const float*)d_in[1];    // teacher_feats [B,D,H,W]
  const float*   C = (const float*)d_in[2];    // center [D]
  const uint8_t* M = (const uint8_t*)d_in[3];  // mask [B,H,W] (bool)
  const float*   O = (const float*)d_in[4];    // original_x [B,1,H,W]
  float* ws  = (float*)d_ws;                   // 2*NBLK floats (32 KB)
  float* out = (float*)d_out;

  dino_pixel_kernel<<<NBLK, 256, 0, stream>>>(S, T, C, M, O, ws);
  dino_finalize_kernel<<<1, 32, 0, stream>>>(ws, out);
}